// WR_Angle_window_42288247996479
// MI455X (gfx1250) — compile-verified
//
#include <hip/hip_runtime.h>

typedef __attribute__((ext_vector_type(2))) float v2f;
typedef __attribute__((ext_vector_type(8))) float v8f;

#define NB      48      // bsz * layers = 4 * 12
#define LAYERS  12
#define MAXLEN  512
#define TDIM    768
#define SDIM    384
#define BANDK   8       // padded stride of the 5-wide dot-product band

// ---------------------------------------------------------------------------
// Stage 1: banded Gram via V_WMMA_F32_16X16X4_F32.
// One wave (32 threads) per (nb-slice, 16-row block).
//   C0[m][n] = dot(e[s0+m], e[s0+n])          (16x16)
//   C1[m][n] = dot(e[s0+m], e[min(s0+16+n,511)])
// Band out: D[i*8+k] = dot(e_i, e_{i+k}), k = 0..4.
// A-fragment layout (ISA 7.12.2, 32-bit A 16x4): lanes 0-15 hold {K0,K1} of
// row M=lane, lanes 16-31 hold {K2,K3} of row M=lane-16. The row-striped
// B-fragment of the transposed tile carries identical per-lane data, so the
// same float2 register feeds both A and B of the symmetric product.
// ---------------------------------------------------------------------------
__global__ __launch_bounds__(32)
void band_gram_wmma(const float* __restrict__ E, float* __restrict__ Dband, int dim)
{
    const int nb   = blockIdx.x >> 5;     // slice
    const int blk  = blockIdx.x & 31;     // 16-row block
    const int s0   = blk << 4;
    const int lane = threadIdx.x;
    const int row  = lane & 15;
    const int kh   = lane >> 4;           // which K-half this lane holds

    const float* base = E + (size_t)nb * MAXLEN * dim;
    const float* aPtr = base + (size_t)(s0 + row) * dim + (kh << 1);
    int rb = s0 + 16 + row; rb = rb > (MAXLEN - 1) ? (MAXLEN - 1) : rb;
    const float* bPtr = base + (size_t)rb * dim + (kh << 1);

    v8f c0 = {0.f,0.f,0.f,0.f,0.f,0.f,0.f,0.f};
    v8f c1 = {0.f,0.f,0.f,0.f,0.f,0.f,0.f,0.f};

#pragma unroll 4
    for (int kc = 0; kc < dim; kc += 4) {
        v2f a = *(const v2f*)(aPtr + kc);   // {E[rowA][kc+2kh], E[rowA][kc+2kh+1]}
        v2f b = *(const v2f*)(bPtr + kc);
        // 8 args: (neg_a, A, neg_b, B, c_mod, C, reuse_a, reuse_b)
        c0 = __builtin_amdgcn_wmma_f32_16x16x4_f32(false, a, false, a, (short)0, c0, false, false);
        c1 = __builtin_amdgcn_wmma_f32_16x16x4_f32(false, a, false, b, (short)0, c1, false, false);
    }

    // C/D layout: VGPR v -> row M = v + 8*(lane>=16), column N = lane%16.
    __shared__ float tile[16][32];
#pragma unroll
    for (int v = 0; v < 8; ++v) {
        int m = v + (kh << 3);
        tile[m][row]      = c0[v];
        tile[m][row + 16] = c1[v];
    }
    __syncthreads();

    float* dout = Dband + ((size_t)nb * MAXLEN + s0) * BANDK;
    for (int t = lane; t < 16 * 5; t += 32) {
        int m = t / 5, k = t - 5 * m;        // column m+k <= 19 < 32
        dout[m * BANDK + k] = tile[m][m + k];
    }
}

// ---------------------------------------------------------------------------
// Stage 2: reconstruct normalized 5x5 Grams from the band, masked SSE.
// ---------------------------------------------------------------------------
__device__ __forceinline__ float band_get(const float* __restrict__ D, int i, int j)
{
    int lo = i < j ? i : j;
    int k  = i < j ? j - i : i - j;          // always <= 4
    return D[lo * BANDK + k];
}

__device__ __forceinline__ void gram5(const float* __restrict__ D, int s,
                                      const int* __restrict__ j, float* __restrict__ g)
{
    float dss = band_get(D, s, s);
    float dsj[5], inv[5];
#pragma unroll
    for (int w = 0; w < 5; ++w) {
        dsj[w] = band_get(D, s, j[w]);
        float djj = band_get(D, j[w], j[w]);
        float nsq = fmaxf(dss - 2.f * dsj[w] + djj, 0.f);
        inv[w] = 1.f / fmaxf(sqrtf(nsq), 1e-12f);   // matches ref: sub/max(||sub||,EPS)
    }
#pragma unroll
    for (int w = 0; w < 5; ++w)
#pragma unroll
        for (int v = 0; v < 5; ++v) {
            float dot = dss - dsj[w] - dsj[v] + band_get(D, j[w], j[v]);
            g[w * 5 + v] = dot * inv[w] * inv[v];
        }
}

__global__ __launch_bounds__(256)
void angle_loss(const float* __restrict__ Dt, const float* __restrict__ Ds,
                const int* __restrict__ amask, float* __restrict__ acc)
{
    int gid = blockIdx.x * 256 + threadIdx.x;
    float num = 0.f, den = 0.f;
    if (gid < NB * MAXLEN) {
        int nb = gid >> 9;
        int s  = gid & (MAXLEN - 1);
        int b  = nb / LAYERS;
        float m = (float)amask[b * MAXLEN + s];

        int j[5];
#pragma unroll
        for (int w = 0; w < 5; ++w) {
            int jj = s - 2 + w;
            j[w] = jj < 0 ? 0 : (jj > MAXLEN - 1 ? MAXLEN - 1 : jj);
        }
        float gt[25], gs[25];
        gram5(Dt + (size_t)nb * MAXLEN * BANDK, s, j, gt);
        gram5(Ds + (size_t)nb * MAXLEN * BANDK, s, j, gs);

        float loc = 0.f;
#pragma unroll
        for (int p = 0; p < 25; ++p) { float d = gs[p] - gt[p]; loc += d * d; }
        num = m * loc;
        den = 25.f * m;     // mask broadcast over the 5x5 window
    }

    __shared__ float red[256];
    int tid = threadIdx.x;
    red[tid] = num; __syncthreads();
    for (int o = 128; o > 0; o >>= 1) { if (tid < o) red[tid] += red[tid + o]; __syncthreads(); }
    if (tid == 0) atomicAdd(&acc[0], red[0]);
    __syncthreads();
    red[tid] = den; __syncthreads();
    for (int o = 128; o > 0; o >>= 1) { if (tid < o) red[tid] += red[tid + o]; __syncthreads(); }
    if (tid == 0) atomicAdd(&acc[1], red[0]);
}

__global__ void zero_acc(float* acc) { acc[0] = 0.f; acc[1] = 0.f; }
__global__ void finalize(const float* acc, float* out) { out[0] = acc[0] / acc[1]; }

// ---------------------------------------------------------------------------
extern "C" void kernel_launch(void* const* d_in, const int* in_sizes, int n_in,
                              void* d_out, int out_size, void* d_ws, size_t ws_size,
                              hipStream_t stream)
{
    (void)in_sizes; (void)n_in; (void)out_size; (void)ws_size;
    const float* t    = (const float*)d_in[0];   // (4,12,512,768) f32
    const float* s    = (const float*)d_in[1];   // (4,12,512,384) f32
    const int*   mask = (const int*)d_in[2];     // (4,512) i32
    // d_in[3] = window (always 5) -- compile-time constant here.

    float* ws  = (float*)d_ws;
    float* acc = ws;                              // [num, den]
    float* Dt  = ws + 64;                         // 256 B offset
    float* Ds  = Dt + (size_t)NB * MAXLEN * BANDK;

    zero_acc<<<1, 1, 0, stream>>>(acc);
    band_gram_wmma<<<NB * 32, 32, 0, stream>>>(t, Dt, TDIM);
    band_gram_wmma<<<NB * 32, 32, 0, stream>>>(s, Ds, SDIM);
    int total = NB * MAXLEN;
    angle_loss<<<(total + 255) / 256, 256, 0, stream>>>(Dt, Ds, mask, acc);
    finalize<<<1, 1, 0, stream>>>(acc, (float*)d_out);
}